// CTCLossWithLabelSmoothing_6914897347094
// MI455X (gfx1250) — compile-verified
//
#include <hip/hip_runtime.h>

// Problem constants (match reference)
#define Tn 2048
#define Bn 64
#define Cn 256
#define Sn 128
#define Ln (2 * Sn + 1)   // 257 extended states
#define NEGF (-1e30f)

#define NBUF 8            // LDS row ring buffer depth (8 KB)
#define DIST 6            // prefetch distance (rows ahead)
#define WAITN 5           // wait asynccnt <= WAITN  => oldest-needed row landed

#ifndef __has_builtin
#define __has_builtin(x) 0
#endif

typedef int v4i __attribute__((ext_vector_type(4)));
typedef __attribute__((address_space(1))) v4i* gptr_v4i;   // global
typedef __attribute__((address_space(3))) v4i* lptr_v4i;   // LDS

// ---- CDNA5 async global->LDS copy (16B per lane) --------------------------
__device__ __forceinline__ void async_b128(const float* gsrc, float* ldst) {
#if __has_builtin(__builtin_amdgcn_global_load_async_to_lds_b128)
  __builtin_amdgcn_global_load_async_to_lds_b128(
      (gptr_v4i)(void*)const_cast<float*>(gsrc),
      (lptr_v4i)(void*)ldst,
      0, 0);
#else
  // VDST = 32-bit LDS byte offset (low 32 bits of the generic LDS address),
  // VADDR = 64-bit global address.  (CDNA5 ISA opcode 98)
  unsigned loff = (unsigned)(unsigned long long)ldst;
  asm volatile("global_load_async_to_lds_b128 %0, %1, off"
               :: "v"(loff), "v"(gsrc) : "memory");
#endif
}

template <int N>
__device__ __forceinline__ void wait_async() {
#if __has_builtin(__builtin_amdgcn_s_wait_asynccnt)
  __builtin_amdgcn_s_wait_asynccnt(N);
#else
  asm volatile("s_wait_asynccnt %0" :: "i"(N) : "memory");
#endif
}

__device__ __forceinline__ float logadd2(float a, float b) {
  float m = fmaxf(a, b);
  return m + __logf(__expf(a - m) + __expf(b - m));
}

// ---- Kernel 1: one workgroup per batch row --------------------------------
__global__ void __launch_bounds__(288)
ctc_alpha_kernel(const float* __restrict__ lp,        // [T,B,C] log-probs
                 const long long* __restrict__ tgt,   // [B,S] int64 targets
                 const int* __restrict__ ilen,        // [B]
                 const int* __restrict__ tlen,        // [B]
                 float* __restrict__ wsLoss,          // [B] loss_b / TL_b
                 float* __restrict__ wsSum)           // [B] sum of log-probs
{
  __shared__ float rowbuf[NBUF][Cn];      // async-filled ring of 1KB rows
  __shared__ float alphaBuf[2][Ln + 2];   // ping-pong alpha, 2-elem NEG pad
  __shared__ float red[256];

  const int b   = blockIdx.x;
  const int tid = threadIdx.x;
  const int TL  = tlen[b];
  const int IL  = ilen[b];
  const int endIdx = 2 * TL;

  // Per-state constants (thread tid handles extended state l = tid)
  int  extv   = 0;      // extended label class (blank = 0 on even states)
  bool skipOk = false;  // may take alpha[l-2] transition
  bool validL = false;
  if (tid < Ln) {
    validL = tid < 2 * TL + 1;
    if (tid & 1) {
      int k = (tid - 1) >> 1;
      extv = (int)tgt[(size_t)b * Sn + k];
      if (tid >= 3) {
        int prev = (int)tgt[(size_t)b * Sn + k - 1];
        skipOk = (extv != 0) && (extv != prev);
      }
    }
  }
  if (tid < 2) { alphaBuf[0][tid] = NEGF; alphaBuf[1][tid] = NEGF; }

  const float* base = lp + (size_t)b * Cn;       // row t at base + t*rstride
  const size_t rstride = (size_t)Bn * Cn;

  // Prologue: issue rows 0..DIST-1 (threads 0..63 = waves 0,1; 16B each)
  if (tid < 64) {
    #pragma unroll
    for (int r = 0; r < DIST; ++r)
      async_b128(base + (size_t)r * rstride + tid * 4, &rowbuf[r][tid * 4]);
  }
  wait_async<WAITN>();       // row 0 landed (in-order completion)
  __syncthreads();

  float accSmooth = 0.0f;
  if (tid < Cn) accSmooth += rowbuf[0][tid];
  if (tid < Ln) {
    float v = NEGF;
    if (tid < 2 && validL) v = rowbuf[0][extv];
    alphaBuf[0][tid + 2] = v;
  }
  if (tid < 64)
    async_b128(base + (size_t)DIST * rstride + tid * 4, &rowbuf[DIST][tid * 4]);
  wait_async<WAITN>();       // rows 0,1 landed
  __syncthreads();

  float finalVal = NEGF;
  if (tid == 0 && IL == 1) {
    float aE = alphaBuf[0][endIdx + 2];
    float aL = alphaBuf[0][(endIdx > 0 ? endIdx - 1 : 0) + 2];
    finalVal = logadd2(aE, aL);
  }

  int cur = 0;
  for (int t = 1; t < Tn; ++t) {
    const float* row = rowbuf[t & (NBUF - 1)];
    const float* A   = alphaBuf[cur];
    float*       Bv  = alphaBuf[cur ^ 1];

    if (tid < Cn) accSmooth += row[tid];        // fused label-smoothing sum
    if (tid < Ln) {
      float lpv = row[extv];                    // LDS gather at ext label
      float a1 = A[tid + 2];
      float a2 = A[tid + 1];
      float a3 = skipOk ? A[tid] : NEGF;
      float m  = fmaxf(a1, fmaxf(a2, a3));
      float nv = m + __logf(__expf(a1 - m) + __expf(a2 - m) + __expf(a3 - m)) + lpv;
      Bv[tid + 2] = validL ? nv : NEGF;
    }
    // Prefetch row t+DIST into the slot last read at step t-2 (2 barriers ago)
    if (tid < 64 && (t + DIST) < Tn)
      async_b128(base + (size_t)(t + DIST) * rstride + tid * 4,
                 &rowbuf[(t + DIST) & (NBUF - 1)][tid * 4]);
    wait_async<WAITN>();     // guarantees row t+1 is resident for next step
    __syncthreads();
    cur ^= 1;
    if (tid == 0 && t == IL - 1) {
      float aE = alphaBuf[cur][endIdx + 2];
      float aL = alphaBuf[cur][(endIdx > 0 ? endIdx - 1 : 0) + 2];
      finalVal = logadd2(aE, aL);
    }
  }

  // Deterministic in-block tree reduction of the smoothing partial sums
  if (tid < 256) red[tid] = accSmooth;
  __syncthreads();
  for (int s = 128; s > 0; s >>= 1) {
    if (tid < s) red[tid] += red[tid + s];
    __syncthreads();
  }
  if (tid == 0) {
    float loss = -finalVal;
    if (loss > 1e29f) loss = 0.0f;              // zero_infinity
    wsLoss[b] = loss / (float)TL;
    wsSum[b]  = red[0];
  }
}

// ---- Kernel 2: deterministic cross-batch reduction ------------------------
__global__ void ctc_finalize(const float* __restrict__ wsLoss,
                             const float* __restrict__ wsSum,
                             float* __restrict__ out) {
  __shared__ float rl[64];
  __shared__ float rs[64];
  int t = threadIdx.x;
  rl[t] = wsLoss[t];
  rs[t] = wsSum[t];
  __syncthreads();
  for (int s = 32; s > 0; s >>= 1) {
    if (t < s) { rl[t] += rl[t + s]; rs[t] += rs[t + s]; }
    __syncthreads();
  }
  if (t == 0) {
    float ctc    = rl[0] / (float)Bn;                                // mean over batch
    float smooth = -rs[0] / ((float)Tn * (float)Bn * (float)Cn);     // -mean(log_probs)
    out[0] = 0.9f * ctc + 0.1f * smooth;   // CONFIDENCE*ctc + SMOOTHING*smooth
  }
}

extern "C" void kernel_launch(void* const* d_in, const int* in_sizes, int n_in,
                              void* d_out, int out_size, void* d_ws, size_t ws_size,
                              hipStream_t stream) {
  const float*     log_probs      = (const float*)d_in[0];
  const long long* targets        = (const long long*)d_in[1];  // int64
  const int*       input_lengths  = (const int*)d_in[2];
  const int*       target_lengths = (const int*)d_in[3];

  float* wsLoss = (float*)d_ws;        // [64]
  float* wsSum  = wsLoss + Bn;         // [64]

  ctc_alpha_kernel<<<Bn, 288, 0, stream>>>(log_probs, targets, input_lengths,
                                           target_lengths, wsLoss, wsSum);
  ctc_finalize<<<1, 64, 0, stream>>>(wsLoss, wsSum, (float*)d_out);
}